// DepthFFN_77403900609179
// MI455X (gfx1250) — compile-verified
//
#include <hip/hip_runtime.h>
#include <hip/hip_bf16.h>

// Problem constants (from reference)
#define BATCH 2
#define CIMG  32
#define DBINS 120          // D
#define HH    47
#define WW    156
#define HWPIX (HH * WW)    // 7332
#define POOLF 8
#define MAPH  (HH * POOLF) // 376
#define MAPW  (WW * POOLF) // 1248
#define OUT1_ELEMS ((size_t)BATCH * CIMG * DBINS * HWPIX)  // 56,309,760

typedef __attribute__((ext_vector_type(2))) float v2f;
typedef __attribute__((ext_vector_type(8))) float v8f;

// ---------------------------------------------------------------------------
// Kernel 1: per-pixel softmax over 121 depth channels -> probs[b, d<120, hw]
// Three streaming passes over channels (logits are 3.5MB/batch, L2-resident
// after the first pass). Coalesced: thread == pixel, channel stride = HW.
// ---------------------------------------------------------------------------
__global__ void softmax_kernel(const float* __restrict__ logits,
                               float* __restrict__ probs) {
    int idx = blockIdx.x * blockDim.x + threadIdx.x;
    if (idx >= BATCH * HWPIX) return;
    int b  = idx / HWPIX;
    int hw = idx - b * HWPIX;
    const float* L = logits + (size_t)b * (DBINS + 1) * HWPIX + hw;

    float mx = -INFINITY;
    for (int ch = 0; ch < DBINS + 1; ++ch)
        mx = fmaxf(mx, L[(size_t)ch * HWPIX]);
    float s = 0.0f;
    for (int ch = 0; ch < DBINS + 1; ++ch)
        s += __expf(L[(size_t)ch * HWPIX] - mx);
    float inv = 1.0f / s;
    float* P = probs + (size_t)b * DBINS * HWPIX + hw;
    for (int ch = 0; ch < DBINS; ++ch)
        P[(size_t)ch * HWPIX] = __expf(L[(size_t)ch * HWPIX] - mx) * inv;
}

// ---------------------------------------------------------------------------
// Kernel 2: the 450MB stream. grid = (pix4, d, b); thread handles 4 pixels.
// out1[b,c,d,hw] = probs[b,d,hw] * feat[b,c,hw]
// out2[b,c,d,hw] = (bin[b,hw]==d) ? feat[b,c,hw] : 0
// All loads/stores are float4-coalesced (HW = 7332 is a multiple of 4, and
// every row offset is 16B aligned). feat (1.9MB) is L2-resident across its
// 120 d-reuses, so HBM traffic ~= the two output streams.
// ---------------------------------------------------------------------------
__global__ void frustum_kernel(const float* __restrict__ feat,
                               const float* __restrict__ probs,
                               const int* __restrict__ bin,
                               float* __restrict__ out) {
    const int HW4 = HWPIX / 4;  // 1833
    int p4 = blockIdx.x * blockDim.x + threadIdx.x;
    if (p4 >= HW4) return;
    int d = blockIdx.y;
    int b = blockIdx.z;

    float4 pv = ((const float4*)(probs + ((size_t)b * DBINS + d) * HWPIX))[p4];
    int4  bv = ((const int4*)(bin + (size_t)b * HWPIX))[p4];
    float4 tv;
    tv.x = (bv.x == d) ? 1.0f : 0.0f;
    tv.y = (bv.y == d) ? 1.0f : 0.0f;
    tv.z = (bv.z == d) ? 1.0f : 0.0f;
    tv.w = (bv.w == d) ? 1.0f : 0.0f;

    float* out2 = out + OUT1_ELEMS;
    #pragma unroll 4
    for (int c = 0; c < CIMG; ++c) {
        float4 f = ((const float4*)(feat + ((size_t)b * CIMG + c) * HWPIX))[p4];
        float4 r1, r2;
        r1.x = f.x * pv.x; r1.y = f.y * pv.y; r1.z = f.z * pv.z; r1.w = f.w * pv.w;
        r2.x = f.x * tv.x; r2.y = f.y * tv.y; r2.z = f.z * tv.z; r2.w = f.w * tv.w;
        size_t row = (((size_t)b * CIMG + c) * DBINS + d) * HWPIX;
        ((float4*)(out  + row))[p4] = r1;
        ((float4*)(out2 + row))[p4] = r2;
    }
}

// ---------------------------------------------------------------------------
// Kernel 3: sparse avg-pool via WMMA reduction.  Each wave reduces 16 output
// pixels (consecutive w).  sum = A x Ones using 16 chained
// V_WMMA_F32_16X16X4_F32 (K=4 each -> 64 elems per output); a second
// accumulator with the nonzero indicator yields the count.  B = all-ones, so
// its (undocumented) VGPR layout is irrelevant.  A layout (documented):
// lane%16 = output row M, lane/16 selects K={0,1} vs K={2,3}, VGPR0=K even,
// VGPR1=K odd.  D column N=0 lives in lanes 0 (M=0..7) and 16 (M=8..15),
// VGPRs 0..7.  EXEC stays all-ones through the WMMA region (clamped indices).
// pooled = sum / (count + 64e-10)   [ == (sum/64)/(cnt/64 + 1e-10) ]
// ---------------------------------------------------------------------------
__global__ void pool_wmma_kernel(const float* __restrict__ maps,
                                 float* __restrict__ pooled,
                                 int nWaves) {
    int gtid = blockIdx.x * blockDim.x + threadIdx.x;
    int wave = gtid >> 5;
    int lane = threadIdx.x & 31;
    bool validWave = wave < nWaves;
    int wv = validWave ? wave : (nWaves - 1);   // clamp: keep EXEC uniform

    const int wavesPerRow = (WW + 15) / 16;     // 10
    int row = wv / wavesPerRow;                 // b*HH + h, 0..93
    int w0  = (wv - row * wavesPerRow) * 16;    // 0,16,...,144
    int b   = row / HH;
    int h   = row - b * HH;

    int m     = lane & 15;   // output column handled by this lane
    int khalf = lane >> 4;   // which pair of K slots this lane feeds
    int w = w0 + m; if (w > WW - 1) w = WW - 1; // clamp (stores masked later)

    const float* base = maps + ((size_t)b * MAPH + (size_t)h * POOLF) * MAPW
                             + (size_t)w * POOLF;

    v8f csum = {};
    v8f ccnt = {};
    v2f ones; ones.x = 1.0f; ones.y = 1.0f;

    #pragma unroll
    for (int k = 0; k < 16; ++k) {
        int e = 4 * k + 2 * khalf;          // flat element index in 8x8 block
        int i = e >> 3;                     // block row
        int j = e & 7;                      // block col (even -> float2 ok)
        float2 dv = *(const float2*)(base + (size_t)i * MAPW + j);
        v2f a;  a.x  = dv.x;  a.y = dv.y;
        v2f ac; ac.x = (dv.x != 0.0f) ? 1.0f : 0.0f;
                ac.y = (dv.y != 0.0f) ? 1.0f : 0.0f;
        csum = __builtin_amdgcn_wmma_f32_16x16x4_f32(
            false, a,  false, ones, (short)0, csum, false, false);
        ccnt = __builtin_amdgcn_wmma_f32_16x16x4_f32(
            false, ac, false, ones, (short)0, ccnt, false, false);
    }

    // Extract column N=0 and store (divergence allowed after WMMAs).
    if (validWave && (lane == 0 || lane == 16)) {
        int mbase = (lane == 16) ? 8 : 0;
        #pragma unroll
        for (int r = 0; r < 8; ++r) {
            int wm = w0 + mbase + r;
            if (wm < WW) {
                pooled[((size_t)b * HH + h) * WW + wm] =
                    csum[r] / (ccnt[r] + 64e-10f);
            }
        }
    }
}

// ---------------------------------------------------------------------------
extern "C" void kernel_launch(void* const* d_in, const int* in_sizes, int n_in,
                              void* d_out, int out_size, void* d_ws, size_t ws_size,
                              hipStream_t stream) {
    const float* feat   = (const float*)d_in[0];  // (2,32,47,156)
    const float* logits = (const float*)d_in[1];  // (2,121,47,156)
    const float* maps   = (const float*)d_in[2];  // (2,376,1248)
    const int*   bin    = (const int*)d_in[3];    // (2,47,156)
    float* out   = (float*)d_out;
    float* probs = (float*)d_ws;                  // needs 2*120*7332*4 = ~7MB

    (void)in_sizes; (void)n_in; (void)out_size; (void)ws_size;

    // 1) softmax -> probs in workspace
    {
        int n = BATCH * HWPIX;
        softmax_kernel<<<(n + 255) / 256, 256, 0, stream>>>(logits, probs);
    }
    // 2) frustum streams (the 450MB of stores)
    {
        dim3 grid((HWPIX / 4 + 255) / 256, DBINS, BATCH);  // (8,120,2)
        frustum_kernel<<<grid, 256, 0, stream>>>(feat, probs, bin, out);
    }
    // 3) sparse avg pool via WMMA, output appended after the two frustum tensors
    {
        int nWaves = BATCH * HH * ((WW + 15) / 16);        // 940
        int threads = nWaves * 32;
        float* pooled = out + 2 * OUT1_ELEMS;
        pool_wmma_kernel<<<(threads + 255) / 256, 256, 0, stream>>>(maps, pooled, nWaves);
    }
}